// BatchRNNCustom_11879879542307
// MI455X (gfx1250) — compile-verified
//
#include <hip/hip_runtime.h>

// ---------------------------------------------------------------------------
// pack_padded_sequence-style packing for (B,S,D) = (64, 2048, 512) fp32.
// Pure data movement: ~23 us roofline at 23.3 TB/s (256 MiB writes + <=256 MiB
// reads). One wave per destination row; binary search offsets[] to invert the
// scatter so every packed row is written exactly once (data or zeros).
// CDNA5 path: async global<->LDS DMA (ASYNCcnt-tracked
// global_load_async_to_lds_b128 / global_store_async_from_lds_b128), plus
// global_prefetch_b8 and nontemporal stores for the write-once output stream.
// ---------------------------------------------------------------------------

namespace {
constexpr int kB = 64;
constexpr int kS = 2048;
constexpr int kD = 512;
constexpr int kRows = kB * kS;        // 131072 destination rows
constexpr int kRowF4 = kD / 4;        // 128 float4 per 2KB row
constexpr int kRowsPerBlock = 8;      // 8 wave32s per 256-thread block

// int workspace layout (in d_ws)
constexpr int kSortedOff   = 0;           // [kB]   sorted_indices
constexpr int kUnsortedOff = kB;          // [kB]   unsorted_indices
constexpr int kOffsetsOff  = 2 * kB;      // [kS+1] exclusive cumsum of batch_sizes
}  // namespace

// Native 16B vector type (HIP float4 is a struct; builtins need real vectors).
typedef float v4f_t __attribute__((ext_vector_type(4)));

#if defined(__HIP_DEVICE_COMPILE__) &&                                        \
    __has_builtin(__builtin_amdgcn_global_load_async_to_lds_b128) &&          \
    __has_builtin(__builtin_amdgcn_global_store_async_from_lds_b128)
#define PACK_ASYNC_LDS 1
#else
#define PACK_ASYNC_LDS 0
#endif

#if PACK_ASYNC_LDS
// Builtin signature (confirmed by clang diagnostics): (v4i addrspace(1)*,
// v4i addrspace(3)*, imm i32 offset, imm i32 cpol).
typedef int v4i_t __attribute__((vector_size(16)));
typedef __attribute__((address_space(1))) v4i_t* gv4i_ptr;
typedef __attribute__((address_space(3))) v4i_t* lv4i_ptr;
#endif

__device__ __forceinline__ void wait_asynccnt0() {
#if defined(__HIP_DEVICE_COMPILE__)
#if __has_builtin(__builtin_amdgcn_s_wait_asynccnt)
  __builtin_amdgcn_s_wait_asynccnt(0);
#else
  asm volatile("s_wait_asynccnt 0" ::: "memory");
#endif
#endif
}

// ---------------------------------------------------------------------------
// The bandwidth-bound scatter (defined first so the disasm snippet shows it).
// One wave32 per destination row r:
//   r >= total -> write 2KB of zeros (nontemporal)
//   else binary-search t with offsets[t] <= r < offsets[t+1],
//        b = r - offsets[t], src = inputs[sorted[b]][t][:]  -> copy 2KB.
// Lane L moves float4 indices {L, L+32, L+64, L+96}: fully coalesced B128.
// ---------------------------------------------------------------------------
__global__ __launch_bounds__(256) void pack_copy_kernel(
    const float* __restrict__ inputs, float* __restrict__ out,
    const int* __restrict__ ws) {
  const int* __restrict__ sortedIdx = ws + kSortedOff;
  const int* __restrict__ offsets   = ws + kOffsetsOff;

  const int lane = threadIdx.x & 31;
  const int wave = threadIdx.x >> 5;
  const long long r = (long long)blockIdx.x * kRowsPerBlock + wave;
  if (r >= kRows) return;

  v4f_t* __restrict__ dst = reinterpret_cast<v4f_t*>(out) + r * kRowF4;
  const int total = offsets[kS];

  if (r >= total) {  // unfilled packed rows must be zero; write-once -> NT
    const v4f_t z = {0.f, 0.f, 0.f, 0.f};
#pragma unroll
    for (int k = 0; k < 4; ++k)
      __builtin_nontemporal_store(z, dst + lane + 32 * k);
    return;
  }

  // wave-uniform binary search over the non-decreasing offsets[0..S]
  int lo = 0, hi = kS;
  while (hi - lo > 1) {
    const int mid = (lo + hi) >> 1;
    if (offsets[mid] <= (int)r) lo = mid; else hi = mid;
  }
  const int t    = lo;
  const int b    = (int)r - offsets[t];
  const int orig = sortedIdx[b];

  const v4f_t* __restrict__ src =
      reinterpret_cast<const v4f_t*>(inputs) +
      ((long long)orig * kS + t) * kRowF4;

  __builtin_prefetch(src, 0, 3);  // global_prefetch_b8 (probe-confirmed)

#if PACK_ASYNC_LDS
  // CDNA5 async DMA path: HBM -> LDS -> HBM, no VGPR data movement,
  // tracked by ASYNCcnt. 2KB staging per wave (16KB per block).
  __shared__ v4f_t stage[kRowsPerBlock * kRowF4];
  v4f_t* st = stage + wave * kRowF4;
#pragma unroll
  for (int k = 0; k < 4; ++k) {
    __builtin_amdgcn_global_load_async_to_lds_b128(
        (gv4i_ptr)(src + lane + 32 * k), (lv4i_ptr)(st + lane + 32 * k), 0, 0);
  }
  wait_asynccnt0();
#pragma unroll
  for (int k = 0; k < 4; ++k) {
    __builtin_amdgcn_global_store_async_from_lds_b128(
        (gv4i_ptr)(dst + lane + 32 * k), (lv4i_ptr)(st + lane + 32 * k), 0, 0);
  }
  wait_asynccnt0();
#else
  v4f_t v0 = src[lane];
  v4f_t v1 = src[lane + 32];
  v4f_t v2 = src[lane + 64];
  v4f_t v3 = src[lane + 96];
  __builtin_nontemporal_store(v0, dst + lane);
  __builtin_nontemporal_store(v1, dst + lane + 32);
  __builtin_nontemporal_store(v2, dst + lane + 64);
  __builtin_nontemporal_store(v3, dst + lane + 96);
#endif
}

// ---------------------------------------------------------------------------
// Metadata kernel. One block, 256 threads; runtime-negligible vs the copy, so
// unrolling is disabled to keep code size / histogram noise down.
//   - stable descending rank sort of 64 lengths (O(B^2) compares)
//   - batch_sizes[t] = #{b : L[b] > t}   (permutation-invariant, use raw L)
//   - exclusive scan -> offsets[0..S], offsets[S] = total
// ---------------------------------------------------------------------------
__global__ __launch_bounds__(256) void pack_setup_kernel(
    const int* __restrict__ lengths, float* __restrict__ out_meta,
    int* __restrict__ ws) {
  __shared__ int sL[kB];
  __shared__ int partial[256];

  const int tid = threadIdx.x;
  if (tid < kB) sL[tid] = lengths[tid];
  __syncthreads();

  // Stable rank for descending sort: rank[i] = #{L[j]>L[i]} + #{j<i: L[j]==L[i]}
  if (tid < kB) {
    const int Li = sL[tid];
    int rank = 0;
#pragma clang loop unroll(disable)
    for (int j = 0; j < kB; ++j) {
      const int Lj = sL[j];
      rank += (Lj > Li) || (Lj == Li && j < tid);
    }
    ws[kSortedOff + rank]  = tid;   // sorted_indices[rank] = i
    ws[kUnsortedOff + tid] = rank;  // unsorted_indices[i]  = rank
    out_meta[kS + rank]      = (float)tid;
    out_meta[kS + kB + tid]  = (float)rank;
  }

  // batch_sizes for 8 consecutive t per thread + local exclusive prefix
  int localExcl[8];
  int mySum = 0;
#pragma unroll
  for (int k = 0; k < 8; ++k) {
    const int t = tid * 8 + k;
    int c = 0;
#pragma clang loop unroll(disable)
    for (int j = 0; j < kB; ++j) c += (sL[j] > t);
    localExcl[k] = mySum;
    mySum += c;
    out_meta[t] = (float)c;  // batch_sizes as float (d_out dtype)
  }
  partial[tid] = mySum;
  __syncthreads();

  // Hillis-Steele inclusive scan over 256 per-thread sums
#pragma clang loop unroll(disable)
  for (int off = 1; off < 256; off <<= 1) {
    const int v = (tid >= off) ? partial[tid - off] : 0;
    __syncthreads();
    partial[tid] += v;
    __syncthreads();
  }
  const int blockExcl = (tid == 0) ? 0 : partial[tid - 1];
#pragma unroll
  for (int k = 0; k < 8; ++k)
    ws[kOffsetsOff + tid * 8 + k] = blockExcl + localExcl[k];
  if (tid == 255) ws[kOffsetsOff + kS] = partial[255];  // total valid rows
}

extern "C" void kernel_launch(void* const* d_in, const int* in_sizes, int n_in,
                              void* d_out, int out_size, void* d_ws,
                              size_t ws_size, hipStream_t stream) {
  (void)in_sizes; (void)n_in; (void)out_size; (void)ws_size;
  const float* inputs  = (const float*)d_in[0];
  // d_in[1] = input_paddings (unused by the reference computation)
  const int*   lengths = (const int*)d_in[2];

  float* out      = (float*)d_out;
  float* out_meta = out + (size_t)kRows * kD;  // batch_sizes | sorted | unsorted
  int*   ws       = (int*)d_ws;

  pack_setup_kernel<<<1, 256, 0, stream>>>(lengths, out_meta, ws);

  const int blocks = kRows / kRowsPerBlock;  // 16384
  pack_copy_kernel<<<blocks, 256, 0, stream>>>(inputs, out, ws);
}